// Luong_exp2_85040352461298
// MI455X (gfx1250) — compile-verified
//
#include <hip/hip_runtime.h>
#include <hip/hip_bf16.h>

// Problem constants (match reference)
#define BATCH 32
#define TENC  1024
#define TDEC  128
#define DENC  512
#define DDEC  256
#define HDIM  256
#define NGATE 1024           // 4*H
#define KCAT  1024           // DENC + DDEC + HDIM
#define CTX_TOTAL (BATCH * TDEC * DENC)   // 2097152 floats, then e_seq follows

typedef __attribute__((ext_vector_type(2))) float v2f;
typedef __attribute__((ext_vector_type(8))) float v8f;

#define WMMA_F32(A, B, C) \
    __builtin_amdgcn_wmma_f32_16x16x4_f32(false, (A), false, (B), (short)0, (C), false, false)

// ---------------------------------------------------------------------------
// Prep: WaT[n][k] = Wa[k][n]  (transposed so WMMA B-operand loads are b64)
// ---------------------------------------------------------------------------
__global__ void prep_waT(const float* __restrict__ Wa, float* __restrict__ WaT) {
    int idx = blockIdx.x * blockDim.x + threadIdx.x;   // 131072 threads
    int n = idx >> 9, k = idx & (DENC - 1);            // n<256, k<512
    WaT[idx] = Wa[k * HDIM + n];
}

// WcatT[n][k] = [lstm_kernel ; lstm_rec_kernel][k][n]  (transposed concat)
__global__ void prep_wcatT(const float* __restrict__ lk, const float* __restrict__ lrk,
                           float* __restrict__ WcatT) {
    int idx = blockIdx.x * blockDim.x + threadIdx.x;   // 1048576 threads
    int n = idx >> 10, k = idx & (KCAT - 1);           // n<1024, k<1024
    WcatT[idx] = (k < 768) ? lk[k * NGATE + n] : lrk[(k - 768) * NGATE + n];
}

__global__ void prep_state(const float* __restrict__ h0, const float* __restrict__ c0,
                           float* __restrict__ h, float* __restrict__ c,
                           float* __restrict__ xcat) {
    int idx = blockIdx.x * blockDim.x + threadIdx.x;   // 8192 threads
    float hv = h0[idx];
    h[idx] = hv;
    c[idx] = c0[idx];
    int b = idx >> 8, j = idx & 255;
    xcat[b * KCAT + DENC + DDEC + j] = hv;             // h slot of xcat
}

// ---------------------------------------------------------------------------
// GEMM-U: U[(b,t), d] = sum_e enc[(b,t), e] * Wa[e, d]
// M = 32768, K = 512, N = 256.  fp32 WMMA 16x16x4, 2x2 register blocking:
// each wave computes a 32x32 tile (4 accumulators), sharing A/B loads 2-way.
// grid = (1024, 2), block = 128 (4 waves, one 32-wide n-pair each).
// ---------------------------------------------------------------------------
__global__ void gemm_U(const float* __restrict__ enc, const float* __restrict__ WaT,
                       float* __restrict__ U) {
    const int wave = threadIdx.x >> 5;
    const int lane = threadIdx.x & 31;
    const int m0 = blockIdx.x * 32;                    // 0..32736
    const int n0 = (blockIdx.y * 4 + wave) * 32;       // 0..224
    const int row  = lane & 15;
    const int koff = (lane >> 4) * 2;                  // 0 or 2
    const float* A0 = enc + (size_t)(m0 + row) * DENC + koff;
    const float* A1 = A0 + (size_t)16 * DENC;
    const float* B0 = WaT + (size_t)(n0 + row) * DENC + koff;
    const float* B1 = B0 + (size_t)16 * DENC;

    v8f acc00 = {}, acc01 = {}, acc10 = {}, acc11 = {};
    #pragma unroll 2
    for (int k = 0; k < DENC; k += 16) {
        __builtin_prefetch(A0 + k + 128, 0, 3);
        __builtin_prefetch(B0 + k + 128, 0, 3);
        #pragma unroll
        for (int kk = 0; kk < 16; kk += 4) {
            v2f a0 = *(const v2f*)(A0 + k + kk);
            v2f a1 = *(const v2f*)(A1 + k + kk);
            v2f b0 = *(const v2f*)(B0 + k + kk);
            v2f b1 = *(const v2f*)(B1 + k + kk);
            acc00 = WMMA_F32(a0, b0, acc00);
            acc01 = WMMA_F32(a0, b1, acc01);
            acc10 = WMMA_F32(a1, b0, acc10);
            acc11 = WMMA_F32(a1, b1, acc11);
        }
    }
    // C/D layout: VGPR r -> lanes 0-15: M=r, lanes 16-31: M=r+8; N = lane&15
    const int col   = lane & 15;
    const int mhalf = (lane >> 4) << 3;
    #pragma unroll
    for (int mi = 0; mi < 2; ++mi) {
        #pragma unroll
        for (int ni = 0; ni < 2; ++ni) {
            v8f acc = (mi == 0) ? (ni == 0 ? acc00 : acc01)
                                : (ni == 0 ? acc10 : acc11);
            float* Urow = U + (size_t)(m0 + mi * 16 + mhalf) * HDIM + n0 + ni * 16 + col;
            #pragma unroll
            for (int r = 0; r < 8; ++r)
                Urow[(size_t)r * HDIM] = acc[r];
        }
    }
}

// ---------------------------------------------------------------------------
// Per-step attention: scores -> softmax -> ctx; writes e/ctx outputs and xcat.
// grid = 32 (one block per batch), block = 1024.
// h broadcast into LDS via gfx1250 async global->LDS load (ASYNCcnt tracked).
// ---------------------------------------------------------------------------
__global__ void attn_step(const float* __restrict__ U, const float* __restrict__ enc,
                          const float* __restrict__ dec, const float* __restrict__ h,
                          float* __restrict__ xcat, float* __restrict__ dout, int step) {
    const int b   = blockIdx.x;
    const int tid = threadIdx.x;
    __shared__ float sh_h[HDIM];
    __shared__ float sh_e[TENC];
    __shared__ float sh_r[TENC];

    if (tid < HDIM) {
        // GLOBAL_LOAD_ASYNC_TO_LDS_B32: copies a global dword into LDS at vdst; tracked by ASYNCcnt
        unsigned lds_off = (unsigned)(size_t)(&sh_h[tid]);
        const float* src = h + b * HDIM + tid;
        asm volatile("global_load_async_to_lds_b32 %0, %1, off"
                     :: "v"(lds_off), "v"(src) : "memory");
        asm volatile("s_wait_asynccnt 0x0" ::: "memory");
    }
    __syncthreads();

    // score for encoder position t = tid: dot(U[b,t,:], h[b,:]) over 256
    const float4* U4 = (const float4*)(U + ((size_t)b * TENC + tid) * HDIM);
    float s = 0.f;
    #pragma unroll 8
    for (int d = 0; d < HDIM / 4; ++d) {
        float4 u = U4[d];
        s = fmaf(u.x, sh_h[4 * d + 0], s);
        s = fmaf(u.y, sh_h[4 * d + 1], s);
        s = fmaf(u.z, sh_h[4 * d + 2], s);
        s = fmaf(u.w, sh_h[4 * d + 3], s);
    }

    // softmax over T_ENC = 1024 within the block
    sh_r[tid] = s;
    __syncthreads();
    for (int off = 512; off > 0; off >>= 1) {
        if (tid < off) sh_r[tid] = fmaxf(sh_r[tid], sh_r[tid + off]);
        __syncthreads();
    }
    const float mx = sh_r[0];
    __syncthreads();
    const float p = __expf(s - mx);
    sh_r[tid] = p;
    __syncthreads();
    for (int off = 512; off > 0; off >>= 1) {
        if (tid < off) sh_r[tid] += sh_r[tid + off];
        __syncthreads();
    }
    const float inv = 1.f / sh_r[0];
    __syncthreads();
    const float en = p * inv;
    sh_e[tid] = en;
    dout[CTX_TOTAL + ((size_t)b * TDEC + step) * TENC + tid] = en;   // e_seq output
    __syncthreads();

    // ctx[f] = sum_t enc[b,t,f] * e[t] ; 1024 threads = 512 features x 2 halves
    const int f    = tid & (DENC - 1);
    const int half = tid >> 9;
    const float* ep   = sh_e + half * 512;
    const float* encb = enc + ((size_t)b * TENC + half * 512) * DENC + f;
    float acc = 0.f;
    #pragma unroll 4
    for (int t = 0; t < 512; ++t)
        acc = fmaf(encb[(size_t)t * DENC], ep[t], acc);
    sh_r[tid] = acc;
    __syncthreads();
    if (tid < DENC) {
        float ctxv = sh_r[tid] + sh_r[tid + DENC];
        xcat[b * KCAT + tid] = ctxv;                                  // ctx slot
        dout[((size_t)b * TDEC + step) * DENC + tid] = ctxv;          // ctx_seq output
    } else if (tid < DENC + DDEC) {
        int j = tid - DENC;                                           // x_t slot
        xcat[b * KCAT + DENC + j] = dec[((size_t)b * TDEC + step) * DDEC + j];
    }
}

// ---------------------------------------------------------------------------
// GEMM-Z: z = xcat @ Wcat + bias.  M = 32, N = 1024, K = 1024 (fp32 WMMA).
// 2x2 register blocking: each wave computes all 32 rows x 32 cols.
// 32 waves total: grid = 8 blocks x 128 threads.
// ---------------------------------------------------------------------------
__global__ void gemm_Z(const float* __restrict__ xcat, const float* __restrict__ WcatT,
                       const float* __restrict__ bias, float* __restrict__ z) {
    const int wave = threadIdx.x >> 5;
    const int lane = threadIdx.x & 31;
    const int n0 = (blockIdx.x * 4 + wave) * 32;       // 0..992
    const int row  = lane & 15;
    const int koff = (lane >> 4) * 2;
    const float* A0 = xcat + (row) * KCAT + koff;          // rows 0..15
    const float* A1 = A0 + 16 * KCAT;                      // rows 16..31
    const float* B0 = WcatT + (n0 + row) * KCAT + koff;
    const float* B1 = B0 + 16 * KCAT;

    v8f acc00 = {}, acc01 = {}, acc10 = {}, acc11 = {};
    #pragma unroll 2
    for (int k = 0; k < KCAT; k += 16) {
        __builtin_prefetch(B0 + k + 128, 0, 3);
        __builtin_prefetch(B1 + k + 128, 0, 3);
        #pragma unroll
        for (int kk = 0; kk < 16; kk += 4) {
            v2f a0 = *(const v2f*)(A0 + k + kk);
            v2f a1 = *(const v2f*)(A1 + k + kk);
            v2f b0 = *(const v2f*)(B0 + k + kk);
            v2f b1 = *(const v2f*)(B1 + k + kk);
            acc00 = WMMA_F32(a0, b0, acc00);
            acc01 = WMMA_F32(a0, b1, acc01);
            acc10 = WMMA_F32(a1, b0, acc10);
            acc11 = WMMA_F32(a1, b1, acc11);
        }
    }
    const int col   = lane & 15;
    const int mhalf = (lane >> 4) << 3;
    #pragma unroll
    for (int mi = 0; mi < 2; ++mi) {
        #pragma unroll
        for (int ni = 0; ni < 2; ++ni) {
            v8f acc = (mi == 0) ? (ni == 0 ? acc00 : acc01)
                                : (ni == 0 ? acc10 : acc11);
            const float bv = bias[n0 + ni * 16 + col];
            float* zrow = z + (mi * 16 + mhalf) * NGATE + n0 + ni * 16 + col;
            #pragma unroll
            for (int r = 0; r < 8; ++r)
                zrow[r * NGATE] = acc[r] + bv;
        }
    }
}

// ---------------------------------------------------------------------------
// LSTM gates: keras order i,f,g,o ; update c,h ; refresh h slot of xcat.
// grid = 64 x 128 = 8192 threads.
// ---------------------------------------------------------------------------
__global__ void lstm_gates(const float* __restrict__ z, float* __restrict__ c,
                           float* __restrict__ h, float* __restrict__ xcat) {
    const int idx = blockIdx.x * blockDim.x + threadIdx.x;   // 0..8191
    const int b = idx >> 8, j = idx & 255;
    const float* zb = z + b * NGATE;
    const float zi = zb[j], zf = zb[256 + j], zg = zb[512 + j], zo = zb[768 + j];
    const float si = 1.f / (1.f + __expf(-zi));
    const float sf = 1.f / (1.f + __expf(-zf));
    const float so = 1.f / (1.f + __expf(-zo));
    const float cn = sf * c[idx] + si * tanhf(zg);
    const float hn = so * tanhf(cn);
    c[idx] = cn;
    h[idx] = hn;
    xcat[b * KCAT + DENC + DDEC + j] = hn;
}

// ---------------------------------------------------------------------------
extern "C" void kernel_launch(void* const* d_in, const int* in_sizes, int n_in,
                              void* d_out, int out_size, void* d_ws, size_t ws_size,
                              hipStream_t stream) {
    const float* enc = (const float*)d_in[0];   // [32,1024,512]
    const float* dec = (const float*)d_in[1];   // [32,128,256]
    const float* h0  = (const float*)d_in[2];   // [32,256]
    const float* c0  = (const float*)d_in[3];   // [32,256]
    const float* Wa  = (const float*)d_in[4];   // [512,256]
    const float* lk  = (const float*)d_in[5];   // [768,1024]
    const float* lrk = (const float*)d_in[6];   // [256,1024]
    const float* lb  = (const float*)d_in[7];   // [1024]
    float* out = (float*)d_out;

    // Workspace carve-up (floats): ~38.6 MB total
    float* ws    = (float*)d_ws;
    float* U     = ws;                                  // 32*1024*256 = 8388608
    float* WaT   = U + (size_t)BATCH * TENC * HDIM;     // 256*512   = 131072
    float* WcatT = WaT + (size_t)HDIM * DENC;           // 1024*1024 = 1048576
    float* xcat  = WcatT + (size_t)NGATE * KCAT;        // 32*1024
    float* z     = xcat + BATCH * KCAT;                 // 32*1024
    float* h     = z + BATCH * NGATE;                   // 32*256
    float* c     = h + BATCH * HDIM;                    // 32*256

    prep_waT<<<(HDIM * DENC) / 256, 256, 0, stream>>>(Wa, WaT);
    prep_wcatT<<<(NGATE * KCAT) / 256, 256, 0, stream>>>(lk, lrk, WcatT);
    prep_state<<<(BATCH * HDIM) / 256, 256, 0, stream>>>(h0, c0, h, c, xcat);

    // Encoder projection: one big fp32 WMMA GEMM (2x2 blocked waves)
    gemm_U<<<dim3((BATCH * TENC) / 32, 2), 128, 0, stream>>>(enc, WaT, U);

    // Sequential decoder loop (true data dependence through h)
    for (int step = 0; step < TDEC; ++step) {
        attn_step<<<BATCH, 1024, 0, stream>>>(U, enc, dec, h, xcat, out, step);
        gemm_Z<<<8, 128, 0, stream>>>(xcat, WcatT, lb, z);
        lstm_gates<<<64, 128, 0, stream>>>(z, c, h, xcat);
    }
}